// InverseHaarTransform_55344948576885
// MI455X (gfx1250) — compile-verified
//
#include <hip/hip_runtime.h>

// Inverse Haar (upfirdn: bilinear 2x up + pad(1 top/left) + 2x2 depthwise conv, summed over 4 bands)
// x: [16,12,512,512] f32 -> out: [16,3,1024,1024] f32.
//
// Strategy (bandwidth-bound: ~402 MB traffic ~= 17us @ 23.3 TB/s):
//  - One TDM tensor_load_to_lds per workgroup pulls the 4-band 34x34 halo tile into LDS (3D tile).
//  - Phase A: combine bands with the 2x2 filter taps ONCE per input pixel:
//        g_ab = f_ll[a,b]*LL + f_lh[a,b]*LH + f_hl[a,b]*HL + f_hh[a,b]*HH   (4 fields in LDS)
//  - Phase B: out[m,n] = up(g00)[m-1,n-1] + up(g01)[m-1,n] + up(g10)[m,n-1] + up(g11)[m,n]
//    with separable half-pixel bilinear weights (0.75/0.25 by parity), clamped at edges,
//    and zeroed taps where the upfirdn zero-pad applies (m-1<0 or n-1<0).

#define TILE   32
#define HALO   34
#define GPITCH 37
#define GSZ    (HALO * GPITCH)   // 1258

typedef unsigned int u32x4 __attribute__((ext_vector_type(4)));
typedef int          i32x8 __attribute__((ext_vector_type(8)));
typedef int          i32x4 __attribute__((ext_vector_type(4)));

__global__ __launch_bounds__(256) void ihaar_tdm_kernel(
    const float* __restrict__ x,
    const float* __restrict__ fll, const float* __restrict__ flh,
    const float* __restrict__ fhl, const float* __restrict__ fhh,
    float* __restrict__ out)
{
    __shared__ float ldsIn[4 * HALO * HALO]; // raw band tiles (TDM dest, packed pitch=cols)
    __shared__ float gF[4 * GSZ];            // combined fields g00,g01,g10,g11 (pitch 37)

    const int tid   = threadIdx.x;
    const int txt   = blockIdx.x;      // col tile 0..15
    const int tyt   = blockIdx.y;      // row tile 0..15
    const int plane = blockIdx.z;      // 0..47  (b*3 + c)
    const int b = plane / 3, c = plane % 3;
    const long long chanBase = (long long)(b * 12 + c) * (512LL * 512LL);

    const int rbase = tyt * TILE - 1;                    // logical halo origin (may be -1)
    const int cbase = txt * TILE - 1;
    const int r_lo  = rbase < 0 ? 0 : rbase;
    const int c_lo  = cbase < 0 ? 0 : cbase;
    const int r_hi  = min(tyt * TILE + TILE, 511);
    const int c_hi  = min(txt * TILE + TILE, 511);
    const int rows  = r_hi - r_lo + 1;                   // 33 or 34
    const int cols  = c_hi - c_lo + 1;
    const int rs0   = r_lo - rbase;                      // 0 or 1
    const int cs0   = c_lo - cbase;
    const int bs    = rows * cols;

    // ---------------- Load 4 band halo tiles into LDS ----------------
#if __has_builtin(__builtin_amdgcn_tensor_load_to_lds)
    if (tid < 32) {  // wave-uniform: only wave 0 issues the DMA
        unsigned long long ga =
            (unsigned long long)(const void*)(x + chanBase + (long long)r_lo * 512 + c_lo);
        unsigned int la = (unsigned int)(unsigned long long)(const void*)&ldsIn[0];

        // D# group 0: count=1 | lds_addr | global_addr(57b) | type=2
        u32x4 g0 = {};
        g0[0] = 1u;
        g0[1] = la;
        g0[2] = (unsigned int)(ga & 0xFFFFFFFFull);
        g0[3] = (unsigned int)((ga >> 32) & 0x1FFFFFFull) | (2u << 30);

        // D# group 1: data_size=4B; tensor dims = remaining extent from tile start;
        // tile = cols x rows x 4 bands; y-stride=512, z-stride=3*512*512 elements.
        const unsigned td0 = (unsigned)(512 - c_lo);
        const unsigned td1 = (unsigned)(512 - r_lo);
        const unsigned s1  = 786432u;                    // 3*512*512
        i32x8 g1 = {};
        g1[0] = (int)(2u << 16);                               // data_size=2 (4 bytes)
        g1[1] = (int)((td0 & 0xFFFFu) << 16);                  // tensor_dim0 lo16
        g1[2] = (int)((td0 >> 16) | ((td1 & 0xFFFFu) << 16));  // dim0 hi | dim1 lo
        g1[3] = (int)((td1 >> 16) | ((unsigned)cols << 16));   // dim1 hi | tile_dim0
        g1[4] = (int)((unsigned)rows | (4u << 16));            // tile_dim1 | tile_dim2=4
        g1[5] = 512;                                           // tensor_dim0_stride lo32
        g1[6] = (int)((s1 & 0xFFFFu) << 16);                   // dim0_str hi16=0 | dim1_str lo16
        g1[7] = (int)(s1 >> 16);                               // tensor_dim1_stride [47:16]

        i32x4 g2 = {}; g2[0] = 4;                              // tensor_dim2 = 4 bands
        i32x4 g3 = {};
        i32x8 g4 = {};                                         // trailing group (unused, zero)

        __builtin_amdgcn_tensor_load_to_lds(g0, g1, g2, g3, g4, 0);
        __builtin_amdgcn_s_wait_tensorcnt(0);
    }
#else
    for (int idx = tid; idx < 4 * HALO * HALO; idx += 256) {
        int band = idx / (HALO * HALO);
        int rem  = idx - band * (HALO * HALO);
        int y = rem / HALO, xx = rem - y * HALO;
        if (y < rows && xx < cols)
            ldsIn[band * bs + y * cols + xx] =
                x[chanBase + (long long)band * 786432LL +
                  (long long)(r_lo + y) * 512 + (c_lo + xx)];
    }
#endif
    __syncthreads();

    // ---------------- Phase A: band combine -> g fields ----------------
    float Fw[4][4];
#pragma unroll
    for (int f = 0; f < 4; ++f) {
        Fw[f][0] = fll[f]; Fw[f][1] = flh[f]; Fw[f][2] = fhl[f]; Fw[f][3] = fhh[f];
    }
    for (int idx = tid; idx < HALO * HALO; idx += 256) {
        int y = idx / HALO, xx = idx - y * HALO;
        if (y < rows && xx < cols) {
            int o = y * cols + xx;
            float v0 = ldsIn[o], v1 = ldsIn[bs + o], v2 = ldsIn[2 * bs + o], v3 = ldsIn[3 * bs + o];
            int so = (y + rs0) * GPITCH + (xx + cs0);
#pragma unroll
            for (int f = 0; f < 4; ++f)
                gF[f * GSZ + so] = Fw[f][0] * v0 + Fw[f][1] * v1 + Fw[f][2] * v2 + Fw[f][3] * v3;
        }
    }
    __syncthreads();

    // ---------------- Phase B: separable upsample stencil ----------------
    // Thread -> one output row segment: row M (64 rows/tile), 16 consecutive cols.
    const int mloc = tid >> 2;
    const int M    = tyt * 64 + mloc;           // global output row
    const int gi   = M >> 1, alpha = M & 1;
    const int N0   = txt * 64 + (tid & 3) * 16; // global output col start
    const int gj0  = N0 >> 1;

    // Row pairs: A = up row M-1, B = up row M (taps on input rows, weights by parity)
    int rA0 = max(gi - 1, 0) - rbase;
    int rA1 = gi - rbase;
    float waA0, waA1, waB0, waB1; int rB0, rB1;
    if (alpha == 0) { waA0 = 0.75f; waA1 = 0.25f; waB0 = 0.25f; waB1 = 0.75f; rB0 = rA0; rB1 = rA1; }
    else            { waA0 = 0.25f; waA1 = 0.75f; waB0 = 0.75f; waB1 = 0.25f; rB0 = rA1; rB1 = min(gi + 1, 511) - rbase; }
    if (M == 0) { waA0 = 0.f; waA1 = 0.f; }     // upfirdn zero-pad: up row -1

    // Row-combined columns: SA uses (g00 w/ A-rows)+(g10 w/ B-rows); SB uses (g01)+(g11).
    float SA[10], SB[10];
#pragma unroll
    for (int q = 0; q < 10; ++q) {
        int gc = gj0 - 1 + q;
        int cq = min(max(gc, 0), 511) - cbase;  // clamped LDS col slot
        float s00 = waA0 * gF[0 * GSZ + rA0 * GPITCH + cq] + waA1 * gF[0 * GSZ + rA1 * GPITCH + cq];
        float s10 = waB0 * gF[2 * GSZ + rB0 * GPITCH + cq] + waB1 * gF[2 * GSZ + rB1 * GPITCH + cq];
        SA[q] = s00 + s10;
        float s01 = waA0 * gF[1 * GSZ + rA0 * GPITCH + cq] + waA1 * gF[1 * GSZ + rA1 * GPITCH + cq];
        float s11 = waB0 * gF[3 * GSZ + rB0 * GPITCH + cq] + waB1 * gF[3 * GSZ + rB1 * GPITCH + cq];
        SB[q] = s01 + s11;
    }

    float res[16];
#pragma unroll
    for (int k = 0; k < 16; ++k) {
        int h = k >> 1, beta = k & 1;
        float a0, a1, b0, b1; int qB;
        if (beta == 0) { a0 = 0.75f; a1 = 0.25f; b0 = 0.25f; b1 = 0.75f; qB = h; }
        else           { a0 = 0.25f; a1 = 0.75f; b0 = 0.75f; b1 = 0.25f; qB = h + 1; }
        if (N0 + k == 0) { a0 = 0.f; a1 = 0.f; }  // upfirdn zero-pad: up col -1
        res[k] = a0 * SA[h] + a1 * SA[h + 1] + b0 * SB[qB] + b1 * SB[qB + 1];
    }

    float4* op = (float4*)(out + (long long)plane * (1024LL * 1024LL) + (long long)M * 1024 + N0);
#pragma unroll
    for (int v = 0; v < 4; ++v)
        op[v] = make_float4(res[4 * v + 0], res[4 * v + 1], res[4 * v + 2], res[4 * v + 3]);
}

extern "C" void kernel_launch(void* const* d_in, const int* in_sizes, int n_in,
                              void* d_out, int out_size, void* d_ws, size_t ws_size,
                              hipStream_t stream) {
    const float* x   = (const float*)d_in[0];
    const float* fll = (const float*)d_in[1];
    const float* flh = (const float*)d_in[2];
    const float* fhl = (const float*)d_in[3];
    const float* fhh = (const float*)d_in[4];
    float* out = (float*)d_out;

    dim3 grid(16, 16, 48);   // 16x16 tiles of 32x32 input pixels, 48 (batch,channel) planes
    dim3 block(256);
    ihaar_tdm_kernel<<<grid, block, 0, stream>>>(x, fll, flh, fhl, fhh, out);
}